// SoftOptionCritic_41893111005231
// MI455X (gfx1250) — compile-verified
//
#include <hip/hip_runtime.h>

typedef __attribute__((ext_vector_type(2))) float v2f;
typedef __attribute__((ext_vector_type(8))) float v8f;

#define B_SZ 4096
#define OBS 256
#define ACTD 64
#define HDIM 1024
#define NOPT 8
#define DDIM 328
#define PADM 4608            // 4096 + 8*64 worst-case 64-aligned segments
#define MT   (PADM / 64)     // 72 M-tiles for routed GEMMs

__device__ __forceinline__ float wave_sum(float v) {
  #pragma unroll
  for (int off = 16; off > 0; off >>= 1) v += __shfl_xor(v, off, 32);
  return v;
}

// ---------------------------------------------------------------------------
// fp32 WMMA GEMM: Y[M,N] = act(X[M,K] * W[K,N] + bias[N])
// 64x128 (or 64x64) tile per block, V_WMMA_F32_16X16X4_F32.
// - A tile in LDS, padded stride (conflict-free 16-lane row reads).
// - B tile in paired-K LDS layout: element (k,n) -> Bs[k>>1][2n + (k&1)],
//   so each lane's {B[k][n], B[k+1][n]} operand is ONE aligned ds_load_b64.
//   Row stride == 32 mod 64 banks -> the two lane-halves use disjoint banks.
// - Double-buffered LDS, register-staged global loads, global_prefetch for
//   the tile after next (one s_barrier per K-block).
// - Fragment loads software-pipelined one k-step ahead so LDS latency hides
//   behind the 4 WMMAs of the previous step (no dscnt==0 stalls).
// ROUTED: per-64-row-tile weight/bias selection via tile_opt table.
// ---------------------------------------------------------------------------
template <int BN, bool RELU, bool ROUTED>
__global__ void __launch_bounds__(32 * (BN / 16))
gemm_f32(const float* __restrict__ X, const float* __restrict__ W,
         const float* __restrict__ bias, float* __restrict__ Y,
         int M, int N, int K,
         const int* __restrict__ tile_opt, long w_stride, long b_stride) {
  constexpr int BM = 64, BK = 32;
  constexpr int ASTR = BK + 4;          // 36 floats: 16B-aligned, conflict-free
  constexpr int BSTR = 2 * BN + 32;     // paired-K row stride (== 32 mod 64 banks)
  constexpr int NT = 32 * (BN / 16);
  constexpr int AITER = (BM * BK / 4) / NT;          // float4 units per thread
  constexpr int BITER = ((BK / 2) * (BN / 2)) / NT;  // pair units per thread
  __shared__ float As[2][BM * ASTR];
  __shared__ float Bs[2][(BK / 2) * BSTR];

  if (ROUTED) {
    int o = tile_opt[blockIdx.y];
    if (o < 0) return;                 // pad tile: no work
    W += (long)o * w_stride;
    bias += (long)o * b_stride;
  }
  const int t = threadIdx.x;
  const int lane = t & 31;
  const int wave = t >> 5;
  const int ml = lane & 15;            // M (for A) / N (for B,C) lane index
  const int kh = lane >> 4;            // lane-half selects K pair / M half
  const int row_off = (wave & 3) * 16;
  const int col_off = (wave >> 2) * 64;
  const int rowBase = blockIdx.y * BM;
  const int colBase = blockIdx.x * BN;

  v8f acc[4];
  #pragma unroll
  for (int j = 0; j < 4; ++j)
    #pragma unroll
    for (int i = 0; i < 8; ++i) acc[j][i] = 0.0f;

  float4 ra[AITER];
  float2 rb0[BITER], rb1[BITER];

  // stage tile at k-offset kb from global into registers
  auto gload = [&](int kb) {
    #pragma unroll
    for (int i = 0; i < AITER; ++i) {
      int f = t + i * NT;
      int r = f >> 3, c4 = f & 7;
      int gk = kb + c4 * 4;
      ra[i] = make_float4(0.f, 0.f, 0.f, 0.f);
      if (gk < K) ra[i] = *(const float4*)&X[(size_t)(rowBase + r) * K + gk];
    }
    #pragma unroll
    for (int i = 0; i < BITER; ++i) {
      int u = t + i * NT;
      int p = u / (BN / 2), c2 = u % (BN / 2);
      int gk = kb + 2 * p;
      rb0[i] = make_float2(0.f, 0.f);
      rb1[i] = make_float2(0.f, 0.f);
      if (gk < K) {
        rb0[i] = *(const float2*)&W[(size_t)gk * N + colBase + 2 * c2];
        rb1[i] = *(const float2*)&W[(size_t)(gk + 1) * N + colBase + 2 * c2];
      }
    }
  };
  // write staged registers into LDS buffer `buf`
  auto sstore = [&](int buf) {
    #pragma unroll
    for (int i = 0; i < AITER; ++i) {
      int f = t + i * NT;
      int r = f >> 3, c4 = f & 7;
      *(float4*)&As[buf][r * ASTR + c4 * 4] = ra[i];
    }
    #pragma unroll
    for (int i = 0; i < BITER; ++i) {
      int u = t + i * NT;
      int p = u / (BN / 2), c2 = u % (BN / 2);
      float4 v = make_float4(rb0[i].x, rb1[i].x, rb0[i].y, rb1[i].y);
      *(float4*)&Bs[buf][p * BSTR + 4 * c2] = v;
    }
  };
  // load one k-step's A/B fragments from LDS into registers
  auto ldfrag = [&](int ks, int buf, float2& af, float2 (&bf)[4]) {
    int k0 = ks * 4 + kh * 2;
    af = *(const float2*)&As[buf][(row_off + ml) * ASTR + k0];
    int p = k0 >> 1;
    #pragma unroll
    for (int j = 0; j < 4; ++j) {
      int cc = col_off + j * 16 + ml;
      bf[j] = *(const float2*)&Bs[buf][p * BSTR + 2 * cc];
    }
  };
  // 4 WMMAs on pre-loaded fragments
  auto cstep = [&](const float2& af, const float2 (&bf)[4]) {
    v2f a; a.x = af.x; a.y = af.y;     // A 16x4: lanes0-15 K={0,1}, 16-31 K={2,3}
    #pragma unroll
    for (int j = 0; j < 4; ++j) {
      v2f b; b.x = bf[j].x; b.y = bf[j].y;
      acc[j] = __builtin_amdgcn_wmma_f32_16x16x4_f32(
          false, a, false, b, (short)0, acc[j], false, false);
    }
  };

  const int nkb = (K + BK - 1) / BK;
  gload(0);
  sstore(0);
  __syncthreads();
  for (int ib = 0; ib < nkb; ++ib) {
    const int kb = ib * BK;
    if (ib + 1 < nkb) {
      gload(kb + BK);                  // overlap next tile with compute
      if (ib + 2 < nkb) {              // warm L2 two tiles ahead
        int kb2 = kb + 2 * BK;
        __builtin_prefetch(&X[(size_t)(rowBase + (t >> 3)) * K + kb2 + (t & 7) * 4], 0, 1);
        __builtin_prefetch(
            &W[(size_t)(kb2 + 2 * (t / (BN / 2))) * N + colBase + 2 * (t % (BN / 2))], 0, 1);
      }
    }
    int kcnt = K - kb; if (kcnt > BK) kcnt = BK;
    const int buf = ib & 1;
    if (kcnt == BK) {
      // software-pipelined: fragments for step ks+1 in flight during step ks
      float2 af[2]; float2 bf[2][4];
      ldfrag(0, buf, af[0], bf[0]);
      #pragma unroll
      for (int ks = 0; ks < BK / 4; ++ks) {
        const int cur = ks & 1;
        if (ks + 1 < BK / 4) ldfrag(ks + 1, buf, af[cur ^ 1], bf[cur ^ 1]);
        cstep(af[cur], bf[cur]);
      }
    } else {
      for (int ks = 0; ks < (kcnt >> 2); ++ks) {
        float2 af; float2 bf[4];
        ldfrag(ks, buf, af, bf);
        cstep(af, bf);
      }
    }
    if (ib + 1 < nkb) sstore((ib + 1) & 1);
    __syncthreads();
  }

  // epilogue: bias + optional relu; C/D layout: VGPR v -> M = v + 8*kh, N = ml
  #pragma unroll
  for (int j = 0; j < 4; ++j) {
    int n = colBase + col_off + j * 16 + ml;
    float bv = bias[n];
    #pragma unroll
    for (int v = 0; v < 8; ++v) {
      int m = rowBase + row_off + v + kh * 8;
      float r = acc[j][v] + bv;
      if (RELU) r = fmaxf(r, 0.f);
      Y[(size_t)m * N + n] = r;
    }
  }
}

// Y[M,8] = X[M,K] * W[K,8] + bias[8]  (inter-Q heads)
__global__ void __launch_bounds__(256)
gemv_n8(const float* __restrict__ X, const float* __restrict__ W,
        const float* __restrict__ bias, float* __restrict__ Y, int K) {
  int lane = threadIdx.x & 31;
  int row = blockIdx.x * 8 + (threadIdx.x >> 5);
  float acc[8] = {0.f, 0.f, 0.f, 0.f, 0.f, 0.f, 0.f, 0.f};
  for (int k = lane; k < K; k += 32) {
    float x = X[(size_t)row * K + k];
    const float4* wp = (const float4*)&W[k * 8];
    float4 w0 = wp[0], w1 = wp[1];
    acc[0] += x * w0.x; acc[1] += x * w0.y; acc[2] += x * w0.z; acc[3] += x * w0.w;
    acc[4] += x * w1.x; acc[5] += x * w1.y; acc[6] += x * w1.z; acc[7] += x * w1.w;
  }
  #pragma unroll
  for (int n = 0; n < 8; ++n) acc[n] = wave_sum(acc[n]);
  if (lane == 0) {
    #pragma unroll
    for (int n = 0; n < 8; ++n) Y[(size_t)row * 8 + n] = acc[n] + bias[n];
  }
}

// Y[M] = X[M,K] * w[K] + b[0]  (intra-Q heads)
__global__ void __launch_bounds__(256)
gemv_n1(const float* __restrict__ X, const float* __restrict__ W,
        const float* __restrict__ bias, float* __restrict__ Y, int K) {
  int lane = threadIdx.x & 31;
  int row = blockIdx.x * 8 + (threadIdx.x >> 5);
  float s = 0.f;
  for (int k = lane * 4; k < K; k += 128) {
    float4 x = *(const float4*)&X[(size_t)row * K + k];
    float4 w = *(const float4*)&W[k];
    s += x.x * w.x + x.y * w.y + x.z * w.z + x.w * w.w;
  }
  s = wave_sum(s);
  if (lane == 0) Y[row] = s + bias[0];
}

// routed beta head: blog[m] = X[m,:] . beta_W2[o] + beta_b2[o]
__global__ void __launch_bounds__(256)
gemv_beta(const float* __restrict__ X, const float* __restrict__ W,
          const float* __restrict__ bias, float* __restrict__ Y, int K,
          const int* __restrict__ tile_opt) {
  int lane = threadIdx.x & 31;
  int row = blockIdx.x * 8 + (threadIdx.x >> 5);
  int o = tile_opt[row >> 6];
  if (o < 0) { if (lane == 0) Y[row] = 0.f; return; }
  const float* w = W + (size_t)o * K;
  float s = 0.f;
  for (int k = lane * 4; k < K; k += 128) {
    float4 x = *(const float4*)&X[(size_t)row * K + k];
    float4 wv = *(const float4*)&w[k];
    s += x.x * wv.x + x.y * wv.y + x.z * wv.z + x.w * wv.w;
  }
  s = wave_sum(s);
  if (lane == 0) Y[row] = s + bias[o];
}

// xq = concat(state, action, onehot(option))
__global__ void build_xq(const float* __restrict__ state, const float* __restrict__ action,
                         const int* __restrict__ option, float* __restrict__ xq, int total) {
  int i = blockIdx.x * 256 + threadIdx.x;
  if (i >= total) return;
  int b = i / DDIM, c = i % DDIM;
  float v;
  if (c < OBS) v = state[(size_t)b * OBS + c];
  else if (c < OBS + ACTD) v = action[(size_t)b * ACTD + (c - OBS)];
  else v = (option[b] == (c - OBS - ACTD)) ? 1.f : 0.f;
  xq[i] = v;
}

__global__ void zero_i32(int* p, int n) {
  int i = blockIdx.x * 32 + threadIdx.x;
  if (i < n) p[i] = 0;
}

__global__ void count_k(const int* __restrict__ option, int* counts) {
  int b = blockIdx.x * 256 + threadIdx.x;
  if (b < B_SZ) atomicAdd(&counts[option[b]], 1);
}

// 64-aligned segment offsets + M-tile -> option table (tiny serial work)
__global__ void plan_k(const int* __restrict__ counts, int* __restrict__ off,
                       int* __restrict__ tile_opt) {
  if (threadIdx.x != 0 || blockIdx.x != 0) return;
  int offs[NOPT + 1];
  offs[0] = 0;
  for (int o = 0; o < NOPT; ++o) offs[o + 1] = offs[o] + ((counts[o] + 63) & ~63);
  for (int o = 0; o <= NOPT; ++o) off[o] = offs[o];
  for (int t = 0; t < MT; ++t) {
    int r = t * 64, sel = -1;
    for (int o = 0; o < NOPT; ++o)
      if (r >= offs[o] && r < offs[o + 1]) sel = o;
    tile_opt[t] = sel;
  }
}

// scatter rows of state/next_state into option-sorted, 64-aligned buffers
__global__ void __launch_bounds__(256)
scatter_k(const float* __restrict__ state, const float* __restrict__ next_state,
          const int* __restrict__ option, const int* __restrict__ off,
          int* cursor, int* __restrict__ pos,
          float* __restrict__ xp, float* __restrict__ xnp) {
  int b = blockIdx.x;
  __shared__ int sp;
  if (threadIdx.x == 0) {
    int o = option[b];
    sp = off[o] + atomicAdd(&cursor[o], 1);
    pos[b] = sp;
  }
  __syncthreads();
  int p = sp, c = threadIdx.x;
  xp[(size_t)p * OBS + c] = state[(size_t)b * OBS + c];
  xnp[(size_t)p * OBS + c] = next_state[(size_t)b * OBS + c];
}

// squashed-Gaussian epilogue: actions, logp (gathered via pos[b])
__global__ void __launch_bounds__(256)
policy_final(const float* __restrict__ mu_p, const float* __restrict__ ls_p,
             const float* __restrict__ noise, const int* __restrict__ pos,
             float* __restrict__ act_out, float* __restrict__ logp_out) {
  const float LOG2PI = 1.8378770664093454f;
  const float LN2 = 0.6931471805599453f;
  int lane = threadIdx.x & 31;
  int b = blockIdx.x * 8 + (threadIdx.x >> 5);
  int p = pos[b];
  float lp = 0.f;
  for (int j = lane; j < ACTD; j += 32) {
    float mu = mu_p[(size_t)p * ACTD + j];
    float ls = ls_p[(size_t)p * ACTD + j];
    ls = fminf(fmaxf(ls, -20.f), 2.f);
    float nz = noise[(size_t)b * ACTD + j];
    float pre = mu + expf(ls) * nz;
    lp += -0.5f * nz * nz - ls - 0.5f * LOG2PI;
    float x = -2.f * pre;
    float sp = fmaxf(x, 0.f) + log1pf(expf(-fabsf(x)));  // softplus(-2*pre)
    lp -= 2.f * (LN2 - pre - sp);
    act_out[(size_t)b * ACTD + j] = tanhf(pre);          // ACT_LIMIT = 1
  }
  lp = wave_sum(lp);
  if (lane == 0) logp_out[b] = lp;
}

// TD backup + per-sample loss terms + beta_prob output
__global__ void backup_k(const float* __restrict__ q_on, const float* __restrict__ q_tg,
                         const float* __restrict__ q_intra, const float* __restrict__ blog_p,
                         const int* __restrict__ pos, const int* __restrict__ option,
                         const float* __restrict__ reward, const float* __restrict__ done,
                         const float* __restrict__ logp_out, float* __restrict__ beta_out,
                         float* __restrict__ e_arr, float* __restrict__ piv) {
  int b = blockIdx.x * 256 + threadIdx.x;
  if (b >= B_SZ) return;
  const float* t0 = q_tg;
  const float* t1 = q_tg + (size_t)B_SZ * NOPT;
  const float* o0 = q_on;
  const float* o1 = q_on + (size_t)B_SZ * NOPT;
  int opt = option[b];
  float qcur = fminf(t0[(size_t)b * NOPT + opt], t1[(size_t)b * NOPT + opt]);
  int bn = 0; float bv = -3.4e38f;
  for (int o = 0; o < NOPT; ++o) {
    float v = fminf(o0[(size_t)b * NOPT + o], o1[(size_t)b * NOPT + o]);
    if (v > bv) { bv = v; bn = o; }   // first-max tie-break like argmax
  }
  float qnext = fminf(t0[(size_t)b * NOPT + bn], t1[(size_t)b * NOPT + bn]);
  float bp = 1.f / (1.f + expf(-blog_p[pos[b]]));
  beta_out[b] = bp;
  float bk = reward[b] + 0.99f * (1.f - done[b]) * ((1.f - bp) * qcur + bp * qnext);
  float q1 = q_intra[b], q2 = q_intra[B_SZ + b];
  float d1 = q1 - bk, d2 = q2 - bk;
  e_arr[b] = d1 * d1 + d2 * d2;
  piv[b] = 0.2f * logp_out[b] - fminf(q1, q2);
}

__global__ void __launch_bounds__(256)
reduce_partial(const float* __restrict__ e, const float* __restrict__ p,
               float* __restrict__ pe, float* __restrict__ pp) {
  __shared__ float sE[256], sP[256];
  int t = threadIdx.x;
  int i = blockIdx.x * 256 + t;
  sE[t] = e[i]; sP[t] = p[i];
  __syncthreads();
  for (int s = 128; s > 0; s >>= 1) {
    if (t < s) { sE[t] += sE[t + s]; sP[t] += sP[t + s]; }
    __syncthreads();
  }
  if (t == 0) { pe[blockIdx.x] = sE[0]; pp[blockIdx.x] = sP[0]; }
}

__global__ void reduce_final(const float* __restrict__ pe, const float* __restrict__ pp,
                             float* __restrict__ out) {
  if (threadIdx.x != 0 || blockIdx.x != 0) return;
  float a = 0.f, c = 0.f;
  for (int i = 0; i < 16; ++i) { a += pe[i]; c += pp[i]; }
  out[0] = a / (float)B_SZ;   // loss_intra_q
  out[1] = c / (float)B_SZ;   // loss_intra_pi
}

// ---------------------------------------------------------------------------
extern "C" void kernel_launch(void* const* d_in, const int* in_sizes, int n_in,
                              void* d_out, int out_size, void* d_ws, size_t ws_size,
                              hipStream_t stream) {
  (void)in_sizes; (void)n_in; (void)out_size; (void)ws_size;
  const float* state      = (const float*)d_in[0];
  const float* action     = (const float*)d_in[1];
  const float* next_state = (const float*)d_in[2];
  const float* reward     = (const float*)d_in[3];
  const float* done       = (const float*)d_in[4];
  const float* noise      = (const float*)d_in[5];
  const int*   option     = (const int*)d_in[6];
  const float* interq_W1  = (const float*)d_in[7];
  const float* interq_b1  = (const float*)d_in[8];
  const float* interq_W2  = (const float*)d_in[9];
  const float* interq_b2  = (const float*)d_in[10];
  const float* interq_W3  = (const float*)d_in[11];
  const float* interq_b3  = (const float*)d_in[12];
  const float* tinterq_W1 = (const float*)d_in[13];
  const float* tinterq_b1 = (const float*)d_in[14];
  const float* tinterq_W2 = (const float*)d_in[15];
  const float* tinterq_b2 = (const float*)d_in[16];
  const float* tinterq_W3 = (const float*)d_in[17];
  const float* tinterq_b3 = (const float*)d_in[18];
  const float* intraq_W1  = (const float*)d_in[19];
  const float* intraq_b1  = (const float*)d_in[20];
  const float* intraq_W2  = (const float*)d_in[21];
  const float* intraq_b2  = (const float*)d_in[22];
  const float* intraq_W3  = (const float*)d_in[23];
  const float* intraq_b3  = (const float*)d_in[24];
  const float* pol_W1     = (const float*)d_in[25];
  const float* pol_b1     = (const float*)d_in[26];
  const float* pol_W2     = (const float*)d_in[27];
  const float* pol_b2     = (const float*)d_in[28];
  const float* pol_Wmu    = (const float*)d_in[29];
  const float* pol_bmu    = (const float*)d_in[30];
  const float* pol_Wls    = (const float*)d_in[31];
  const float* pol_bls    = (const float*)d_in[32];
  const float* beta_W1    = (const float*)d_in[33];
  const float* beta_b1    = (const float*)d_in[34];
  const float* beta_W2    = (const float*)d_in[35];
  const float* beta_b2    = (const float*)d_in[36];

  float* out = (float*)d_out;
  float* out_actions = out + 2;
  float* out_logp = out + 2 + (size_t)B_SZ * ACTD;
  float* out_beta = out_logp + B_SZ;

  // workspace carve-up (256B aligned regions)
  char* w = (char*)d_ws;
  auto carve = [&](size_t bytes) -> char* {
    char* r = w;
    w += (bytes + 255) & ~(size_t)255;
    return r;
  };
  float* xq      = (float*)carve(sizeof(float) * (size_t)B_SZ * DDIM);
  float* bigA    = (float*)carve(sizeof(float) * (size_t)PADM * HDIM);  // ping
  float* bigB    = (float*)carve(sizeof(float) * (size_t)PADM * HDIM);  // pong
  float* q_on    = (float*)carve(sizeof(float) * 2 * (size_t)B_SZ * NOPT);
  float* q_tg    = (float*)carve(sizeof(float) * 2 * (size_t)B_SZ * NOPT);
  float* q_intra = (float*)carve(sizeof(float) * 2 * (size_t)B_SZ);
  float* xp      = (float*)carve(sizeof(float) * (size_t)PADM * OBS);
  float* xnp     = (float*)carve(sizeof(float) * (size_t)PADM * OBS);
  float* mu_p    = (float*)carve(sizeof(float) * (size_t)PADM * ACTD);
  float* ls_p    = (float*)carve(sizeof(float) * (size_t)PADM * ACTD);
  float* blog_p  = (float*)carve(sizeof(float) * (size_t)PADM);
  float* e_arr   = (float*)carve(sizeof(float) * (size_t)B_SZ);
  float* piv     = (float*)carve(sizeof(float) * (size_t)B_SZ);
  float* partE   = (float*)carve(sizeof(float) * 16);
  float* partP   = (float*)carve(sizeof(float) * 16);
  int* counts    = (int*)carve(sizeof(int) * 16);   // counts[8] + cursor[8]
  int* cursor    = counts + 8;
  int* off       = (int*)carve(sizeof(int) * (NOPT + 1));
  int* pos       = (int*)carve(sizeof(int) * B_SZ);
  int* tile_opt  = (int*)carve(sizeof(int) * MT);

  // ---- phase 1: online + target inter-Q nets on next_state ----
  for (int z = 0; z < 4; ++z) {
    const float* W1 = (z < 2 ? interq_W1 : tinterq_W1) + (size_t)(z & 1) * OBS * HDIM;
    const float* b1 = (z < 2 ? interq_b1 : tinterq_b1) + (size_t)(z & 1) * HDIM;
    const float* W2 = (z < 2 ? interq_W2 : tinterq_W2) + (size_t)(z & 1) * HDIM * HDIM;
    const float* b2 = (z < 2 ? interq_b2 : tinterq_b2) + (size_t)(z & 1) * HDIM;
    const float* W3 = (z < 2 ? interq_W3 : tinterq_W3) + (size_t)(z & 1) * HDIM * NOPT;
    const float* b3 = (z < 2 ? interq_b3 : tinterq_b3) + (size_t)(z & 1) * NOPT;
    float* qdst = (z < 2 ? q_on : q_tg) + (size_t)(z & 1) * B_SZ * NOPT;
    gemm_f32<128, true, false><<<dim3(8, 64), 256, 0, stream>>>(
        next_state, W1, b1, bigA, B_SZ, HDIM, OBS, nullptr, 0, 0);
    gemm_f32<128, true, false><<<dim3(8, 64), 256, 0, stream>>>(
        bigA, W2, b2, bigB, B_SZ, HDIM, HDIM, nullptr, 0, 0);
    gemv_n8<<<B_SZ / 8, 256, 0, stream>>>(bigB, W3, b3, qdst, HDIM);
  }

  // ---- phase 2: intra-Q on concat(state, action, onehot) ----
  build_xq<<<(B_SZ * DDIM + 255) / 256, 256, 0, stream>>>(state, action, option, xq,
                                                          B_SZ * DDIM);
  for (int z = 0; z < 2; ++z) {
    gemm_f32<128, true, false><<<dim3(8, 64), 256, 0, stream>>>(
        xq, intraq_W1 + (size_t)z * DDIM * HDIM, intraq_b1 + (size_t)z * HDIM, bigA,
        B_SZ, HDIM, DDIM, nullptr, 0, 0);
    gemm_f32<128, true, false><<<dim3(8, 64), 256, 0, stream>>>(
        bigA, intraq_W2 + (size_t)z * HDIM * HDIM, intraq_b2 + (size_t)z * HDIM, bigB,
        B_SZ, HDIM, HDIM, nullptr, 0, 0);
    gemv_n1<<<B_SZ / 8, 256, 0, stream>>>(bigB, intraq_W3 + (size_t)z * HDIM,
                                          intraq_b3 + z, q_intra + (size_t)z * B_SZ, HDIM);
  }

  // ---- phase 3: option routing (sort batch into 64-aligned segments) ----
  zero_i32<<<1, 32, 0, stream>>>(counts, 16);
  count_k<<<16, 256, 0, stream>>>(option, counts);
  plan_k<<<1, 1, 0, stream>>>(counts, off, tile_opt);
  scatter_k<<<B_SZ, 256, 0, stream>>>(state, next_state, option, off, cursor, pos, xp, xnp);

  // ---- phase 4: routed policy MLP (only the selected expert per sample) ----
  gemm_f32<128, true, true><<<dim3(8, MT), 256, 0, stream>>>(
      xp, pol_W1, pol_b1, bigA, PADM, HDIM, OBS, tile_opt, (long)OBS * HDIM, HDIM);
  gemm_f32<128, true, true><<<dim3(8, MT), 256, 0, stream>>>(
      bigA, pol_W2, pol_b2, bigB, PADM, HDIM, HDIM, tile_opt, (long)HDIM * HDIM, HDIM);
  gemm_f32<64, false, true><<<dim3(1, MT), 128, 0, stream>>>(
      bigB, pol_Wmu, pol_bmu, mu_p, PADM, ACTD, HDIM, tile_opt, (long)HDIM * ACTD, ACTD);
  gemm_f32<64, false, true><<<dim3(1, MT), 128, 0, stream>>>(
      bigB, pol_Wls, pol_bls, ls_p, PADM, ACTD, HDIM, tile_opt, (long)HDIM * ACTD, ACTD);

  // ---- phase 5: routed beta net on next_state ----
  gemm_f32<128, true, true><<<dim3(8, MT), 256, 0, stream>>>(
      xnp, beta_W1, beta_b1, bigA, PADM, HDIM, OBS, tile_opt, (long)OBS * HDIM, HDIM);
  gemv_beta<<<PADM / 8, 256, 0, stream>>>(bigA, beta_W2, beta_b2, blog_p, HDIM, tile_opt);

  // ---- phase 6: epilogues + deterministic loss reductions ----
  policy_final<<<B_SZ / 8, 256, 0, stream>>>(mu_p, ls_p, noise, pos, out_actions, out_logp);
  backup_k<<<16, 256, 0, stream>>>(q_on, q_tg, q_intra, blog_p, pos, option, reward, done,
                                   out_logp, out_beta, e_arr, piv);
  reduce_partial<<<16, 256, 0, stream>>>(e_arr, piv, partE, partP);
  reduce_final<<<1, 1, 0, stream>>>(partE, partP, out);
}